// WeightedMAELoss_58969900974611
// MI455X (gfx1250) — compile-verified
//
#include <hip/hip_runtime.h>

// WeightedMAELoss for MI455X (gfx1250, wave32).
// Pass 1: streaming kernel, float4 NT loads, branch-free bin select,
//         per-wave reduction via v_wmma_f32_16x16x4_f32 (B = ones),
//         one partial per block into d_ws.
// Pass 2: single block reduces partials in fixed order -> mean. Deterministic.

typedef __attribute__((ext_vector_type(4))) float f4;
typedef __attribute__((ext_vector_type(2))) float v2f;
typedef __attribute__((ext_vector_type(8))) float v8f;

__device__ __forceinline__ float wave_reduce_sum(float acc) {
#if __has_builtin(__builtin_amdgcn_wmma_f32_16x16x4_f32)
    // A (16x4 f32): lanes 0-15 row m = lane, VGPR0=K0, VGPR1=K1;
    //               lanes 16-31 row m = lane-16, VGPR0=K2, VGPR1=K3.
    // a = {acc, 0} -> A(m,0)=acc_m, A(m,2)=acc_{m+16}, others 0.
    // B = all ones  -> D(m,n) = acc_m + acc_{m+16} for every n.
    v2f a; a[0] = acc; a[1] = 0.0f;
    v2f b; b[0] = 1.0f; b[1] = 1.0f;
    v8f c = {};
    v8f d = __builtin_amdgcn_wmma_f32_16x16x4_f32(
        /*neg_a=*/false, a, /*neg_b=*/false, b,
        /*c_mod=*/(short)0, c, /*reuse_a=*/false, /*reuse_b=*/false);
    // Lane l<16 holds D(r,l) in VGPR r (rows 0..7); lane l>=16 rows 8..15.
    float t = d[0] + d[1] + d[2] + d[3] + d[4] + d[5] + d[6] + d[7];
    t += __shfl_xor(t, 16, 32);   // combine the two half-wave sums
    return t;                     // full 32-lane sum in every lane
#else
    float t = acc;
#pragma unroll
    for (int off = 16; off > 0; off >>= 1) t += __shfl_xor(t, off, 32);
    return t;
#endif
}

// Valid result only on threadIdx.x == 0. Assumes blockDim.x == 256.
__device__ __forceinline__ float block_reduce_sum_256(float acc) {
    __shared__ float ws[8];
    float t = wave_reduce_sum(acc);
    const int lane = threadIdx.x & 31;
    const int wid  = threadIdx.x >> 5;
    if (lane == 0) ws[wid] = t;
    __syncthreads();
    float total = 0.0f;
    if (threadIdx.x == 0) {
#pragma unroll
        for (int i = 0; i < 8; ++i) total += ws[i];
    }
    return total;
}

__device__ __forceinline__ float bin_weight(
    float t,
    float e1, float e2, float e3, float e4, float e5,
    float e6, float e7, float e8, float e9,
    float w0, float w1, float w2, float w3, float w4,
    float w5, float w6, float w7, float w8, float w9) {
    // idx = clip(searchsorted(edges, t, 'right') - 1, 0, 9)
    // == largest k in [0,9] with t >= edges[k] (0 if t < edges[1]... edges[0]).
    float w = w0;
    w = (t >= e1) ? w1 : w;
    w = (t >= e2) ? w2 : w;
    w = (t >= e3) ? w3 : w;
    w = (t >= e4) ? w4 : w;
    w = (t >= e5) ? w5 : w;
    w = (t >= e6) ? w6 : w;
    w = (t >= e7) ? w7 : w;
    w = (t >= e8) ? w8 : w;
    w = (t >= e9) ? w9 : w;
    return w;
}

__global__ __launch_bounds__(256) void wmael_partial_kernel(
    const float* __restrict__ pred, const float* __restrict__ targ,
    const float* __restrict__ edges, const float* __restrict__ wts,
    float* __restrict__ partials, int n) {
    // Uniform pointers + constant indices -> scalar (s_load) broadcasts.
    const float e1 = edges[1], e2 = edges[2], e3 = edges[3], e4 = edges[4];
    const float e5 = edges[5], e6 = edges[6], e7 = edges[7], e8 = edges[8];
    const float e9 = edges[9];
    const float w0 = wts[0], w1 = wts[1], w2 = wts[2], w3 = wts[3];
    const float w4 = wts[4], w5 = wts[5], w6 = wts[6], w7 = wts[7];
    const float w8 = wts[8], w9 = wts[9];

    const int n4     = n >> 2;
    const int gid    = blockIdx.x * 256 + threadIdx.x;
    const int stride = gridDim.x * 256;
    const f4* __restrict__ p4 = (const f4*)pred;
    const f4* __restrict__ t4 = (const f4*)targ;

    float acc = 0.0f;
    for (int i = gid; i < n4; i += stride) {
        f4 p = __builtin_nontemporal_load(p4 + i);
        f4 t = __builtin_nontemporal_load(t4 + i);
#pragma unroll
        for (int k = 0; k < 4; ++k) {
            const float tv = t[k];
            const float w  = bin_weight(tv, e1, e2, e3, e4, e5, e6, e7, e8, e9,
                                        w0, w1, w2, w3, w4, w5, w6, w7, w8, w9);
            acc = fmaf(fabsf(p[k] - tv), w, acc);
        }
    }
    // Scalar tail (n % 4) handled by global thread 0; reconverges before WMMA.
    if (gid == 0) {
        for (int i = n4 << 2; i < n; ++i) {
            const float tv = targ[i];
            const float w  = bin_weight(tv, e1, e2, e3, e4, e5, e6, e7, e8, e9,
                                        w0, w1, w2, w3, w4, w5, w6, w7, w8, w9);
            acc = fmaf(fabsf(pred[i] - tv), w, acc);
        }
    }

    const float total = block_reduce_sum_256(acc);
    if (threadIdx.x == 0) partials[blockIdx.x] = total;
}

__global__ __launch_bounds__(256) void wmael_final_kernel(
    const float* __restrict__ partials, int num_partials,
    float* __restrict__ out, float inv_n) {
    float acc = 0.0f;
    for (int i = threadIdx.x; i < num_partials; i += 256) acc += partials[i];
    const float total = block_reduce_sum_256(acc);
    if (threadIdx.x == 0) out[0] = total * inv_n;
}

extern "C" void kernel_launch(void* const* d_in, const int* in_sizes, int n_in,
                              void* d_out, int out_size, void* d_ws, size_t ws_size,
                              hipStream_t stream) {
    const float* pred  = (const float*)d_in[0];  // (N,1) f32
    const float* targ  = (const float*)d_in[1];  // (N,)  f32
    const float* edges = (const float*)d_in[2];  // (11,) f32
    const float* wts   = (const float*)d_in[3];  // (10,) f32
    float* out      = (float*)d_out;
    float* partials = (float*)d_ws;              // blocks * 4 bytes (<= 8 KB)

    const int n  = in_sizes[1];
    const int n4 = n >> 2;
    int blocks = (n4 + 255) / 256;
    if (blocks > 2048) blocks = 2048;            // 512K threads, 4x float4/thread at N=2^23
    if (blocks < 1)    blocks = 1;

    wmael_partial_kernel<<<dim3(blocks), dim3(256), 0, stream>>>(
        pred, targ, edges, wts, partials, n);
    wmael_final_kernel<<<dim3(1), dim3(256), 0, stream>>>(
        partials, blocks, out, 1.0f / (float)n);
}